// GlobalAttention_14001593385056
// MI455X (gfx1250) — compile-verified
//
#include <hip/hip_runtime.h>
#include <math.h>

// Problem constants (match reference)
constexpr int SL   = 1024;
constexpr int BS   = 64;
constexpr int HID  = 1024;
constexpr int CELL = 1024;
constexpr int DD   = 2 * CELL;        // 2048
constexpr int KK   = DD + HID;        // 3072 (cat width)
constexpr int NSEG = 8;               // split-K segments for context reduction

typedef __attribute__((ext_vector_type(2))) float v2f;
typedef __attribute__((ext_vector_type(8))) float v8f;

// ---------------------------------------------------------------------------
// Kernel 1: scores[s,b] = dot(Wh_s[s,b,:], h_t[b,:])   (one wave per (s,b))
// ---------------------------------------------------------------------------
__global__ void scores_kernel(const float* __restrict__ Wh_s,
                              const float* __restrict__ h_t,
                              float* __restrict__ scores) {
    int wave = (blockIdx.x * blockDim.x + threadIdx.x) >> 5;
    int lane = threadIdx.x & 31;
    int s = wave / BS;
    int b = wave % BS;

    const float* wp = Wh_s + (size_t)(s * BS + b) * HID;
    const float* hp = h_t + b * HID;

    float sum = 0.f;
    for (int k = lane * 4; k < HID; k += 32 * 4) {
        float4 w = *(const float4*)(wp + k);   // global_load_b128
        float4 h = *(const float4*)(hp + k);
        sum += w.x * h.x + w.y * h.y + w.z * h.z + w.w * h.w;
    }
    for (int off = 16; off >= 1; off >>= 1)
        sum += __shfl_down(sum, off, 32);
    if (lane == 0) scores[s * BS + b] = sum;
}

// ---------------------------------------------------------------------------
// Kernel 2: column softmax over s, apply mask, renormalize (+1e-8),
// write alignment back in place and alignment.T to the output.
// One block per batch column b.
// ---------------------------------------------------------------------------
__device__ __forceinline__ float block_reduce(float v, float* red, int tid, bool is_max) {
    red[tid] = v;
    __syncthreads();
    for (int off = 128; off >= 1; off >>= 1) {
        if (tid < off)
            red[tid] = is_max ? fmaxf(red[tid], red[tid + off]) : (red[tid] + red[tid + off]);
        __syncthreads();
    }
    float r = red[0];
    __syncthreads();
    return r;
}

__global__ void softmax_kernel(const float* __restrict__ mask,
                               float* __restrict__ scores,      // in/out -> alignment
                               float* __restrict__ out_alignT) {
    __shared__ float red[256];
    int b = blockIdx.x;
    int tid = threadIdx.x;

    float m = -3.402823466e38f;
    for (int s = tid; s < SL; s += 256) m = fmaxf(m, scores[s * BS + b]);
    float mx = block_reduce(m, red, tid, true);

    float sum = 0.f;
    for (int s = tid; s < SL; s += 256) sum += __expf(scores[s * BS + b] - mx);
    float tot = block_reduce(sum, red, tid, false);
    float inv = 1.f / tot;

    float msum = 0.f;
    for (int s = tid; s < SL; s += 256) {
        float a = __expf(scores[s * BS + b] - mx) * inv * mask[s * BS + b];
        scores[s * BS + b] = a;
        msum += a;
    }
    float mtot = block_reduce(msum, red, tid, false);
    float rinv = 1.f / (mtot + 1e-8f);

    for (int s = tid; s < SL; s += 256) {
        float a = scores[s * BS + b] * rinv;
        scores[s * BS + b] = a;
        out_alignT[b * SL + s] = a;       // alignment.T (bs, sl)
    }
}

// ---------------------------------------------------------------------------
// Kernel 3: split-K partial context sums (deterministic, no atomics).
// grid = (DD/1024, BS, NSEG), block = 256, each thread owns a float4 of d.
// ---------------------------------------------------------------------------
__global__ void context_partial_kernel(const float* __restrict__ alignment,
                                       const float* __restrict__ h_s,
                                       float* __restrict__ partial) {
    int b   = blockIdx.y;
    int seg = blockIdx.z;
    int d   = (blockIdx.x * 256 + threadIdx.x) * 4;
    int s0  = seg * (SL / NSEG);
    int s1  = s0 + (SL / NSEG);

    float4 acc = make_float4(0.f, 0.f, 0.f, 0.f);
    for (int s = s0; s < s1; ++s) {
        float a = alignment[s * BS + b];
        float4 v = *(const float4*)(h_s + (size_t)(s * BS + b) * DD + d);  // b128 stream
        acc.x += a * v.x; acc.y += a * v.y; acc.z += a * v.z; acc.w += a * v.w;
    }
    *(float4*)(partial + (size_t)(b * NSEG + seg) * DD + d) = acc;
}

// ---------------------------------------------------------------------------
// Kernel 4: reduce partials + build cat = [context | h_t]  (64 x 3072)
// ---------------------------------------------------------------------------
__global__ void build_cat_kernel(const float* __restrict__ partial,
                                 const float* __restrict__ h_t,
                                 float* __restrict__ cat) {
    int b = blockIdx.x;
    for (int k = threadIdx.x; k < KK; k += 256) {
        float v;
        if (k < DD) {
            v = 0.f;
            for (int seg = 0; seg < NSEG; ++seg)
                v += partial[(size_t)(b * NSEG + seg) * DD + k];
        } else {
            v = h_t[b * HID + (k - DD)];
        }
        cat[b * KK + k] = v;
    }
}

// ---------------------------------------------------------------------------
// Kernel 5: h_tilde = tanh(cat @ W_hidden^T + b_hidden) via fp32 WMMA.
// D = A(16x4 f32) x B(4x16 f32) + C, K-loop over 3072 in steps of 4,
// two accumulators to break the WMMA->WMMA RAW chain.
// grid = (M/16=4, N/16=64), block = 32 (one wave, EXEC all ones).
// ---------------------------------------------------------------------------
__global__ void gemm_tanh_wmma(const float* __restrict__ cat,
                               const float* __restrict__ W,      // (1024, 3072) row-major
                               const float* __restrict__ bias,
                               float* __restrict__ out) {        // (64, 1024)
    int m0 = blockIdx.x * 16;
    int n0 = blockIdx.y * 16;
    int lane = threadIdx.x;
    int half = lane >> 4;          // 0: K=0,1   1: K=2,3
    int l15  = lane & 15;

    // A: row m0+l15 of cat; B: row n0+l15 of W (B = W^T so K runs along W's row)
    const float* ap = cat + (m0 + l15) * KK + half * 2;
    const float* bp = W   + (n0 + l15) * KK + half * 2;

    v8f c0 = {};
    v8f c1 = {};
    for (int k = 0; k < KK; k += 8) {
        v2f a0, b0, a1, b1;
        a0.x = ap[k];     a0.y = ap[k + 1];
        b0.x = bp[k];     b0.y = bp[k + 1];
        a1.x = ap[k + 4]; a1.y = ap[k + 5];
        b1.x = bp[k + 4]; b1.y = bp[k + 5];
        c0 = __builtin_amdgcn_wmma_f32_16x16x4_f32(false, a0, false, b0,
                                                   (short)0, c0, false, false);
        c1 = __builtin_amdgcn_wmma_f32_16x16x4_f32(false, a1, false, b1,
                                                   (short)0, c1, false, false);
    }

    float bn = bias[n0 + l15];
    #pragma unroll
    for (int j = 0; j < 8; ++j) {
        int row = m0 + j + half * 8;               // C/D layout: VGPRj = M=j (+8 hi lanes)
        float v = c0[j] + c1[j] + bn;
        out[row * CELL + n0 + l15] = tanhf(v);
    }
}

// ---------------------------------------------------------------------------
extern "C" void kernel_launch(void* const* d_in, const int* in_sizes, int n_in,
                              void* d_out, int out_size, void* d_ws, size_t ws_size,
                              hipStream_t stream) {
    const float* Wh_s  = (const float*)d_in[0];
    const float* h_t   = (const float*)d_in[1];
    const float* mask  = (const float*)d_in[2];
    const float* h_s   = (const float*)d_in[3];
    const float* W     = (const float*)d_in[4];
    const float* bias  = (const float*)d_in[5];
    float* out = (float*)d_out;                 // [0,65536): h_tilde, [65536,131072): alignment.T
    float* ws  = (float*)d_ws;

    float* scores  = ws;                                  // SL*BS            = 65536 f
    float* partial = scores + SL * BS;                    // BS*NSEG*DD       = 1048576 f
    float* cat     = partial + (size_t)BS * NSEG * DD;    // BS*KK            = 196608 f

    // 1) scores: one wave per (s,b) -> 65536 waves, 8 waves/block
    scores_kernel<<<(SL * BS) / 8, 256, 0, stream>>>(Wh_s, h_t, scores);

    // 2) softmax over s per column b (+ mask + renorm), emits alignment.T
    softmax_kernel<<<BS, 256, 0, stream>>>(mask, scores, out + BS * CELL);

    // 3) context split-K partials: 512 MB stream of h_s
    dim3 g3(DD / 1024, BS, NSEG);
    context_partial_kernel<<<g3, 256, 0, stream>>>(scores, h_s, partial);

    // 4) reduce partials + concat h_t
    build_cat_kernel<<<BS, 256, 0, stream>>>(partial, h_t, cat);

    // 5) fp32 WMMA GEMM + bias + tanh
    dim3 g5(BS / 16, CELL / 16);
    gemm_tanh_wmma<<<g5, 32, 0, stream>>>(cat, W, bias, out);
}